// GCNwithJK_P_1623497638171
// MI455X (gfx1250) — compile-verified
//
#include <hip/hip_runtime.h>
#include <hip/hip_bf16.h>

typedef float v2f __attribute__((ext_vector_type(2)));
typedef float v8f __attribute__((ext_vector_type(8)));
typedef unsigned int u32x4 __attribute__((ext_vector_type(4)));
typedef int i32x8 __attribute__((ext_vector_type(8)));
typedef int i32x4 __attribute__((ext_vector_type(4)));

#define HIDC 128
#define LSTRIDE 132    // A-tile float stride: 528B row pitch -> banks {4i,4i+1}, conflict-free
#define PSTRIDE 144    // W pair-row stride in v2f: 144*8 % 256 == 128 -> halves hit disjoint banks
#define BN_INV 0.999995000037499687f   // 1/sqrt(1 + 1e-5)

__device__ __forceinline__ void atomAddF(float* p, float v) {
  __hip_atomic_fetch_add(p, v, __ATOMIC_RELAXED, __HIP_MEMORY_SCOPE_AGENT);
}

// ---------------------------------------------------------------------------
// TDM: async-load a 128-wide x 128-row fp32 tile into LDS with +4-DWORD row
// padding (-> 132-float pitch) and hardware zero-fill past rows_avail.
// D# per CDNA5 ISA 8.3/8.4: count=1, type=2(image), data_size=2(4B),
// pad_enable=1, pad_interval=6(128 DW), pad_amount=3(4 DW).
// This toolchain exposes the 6-arg builtin (clang-23 / therock-10 form).
// ---------------------------------------------------------------------------
__device__ __forceinline__ void tdm_load_tile128(const float* gsrc,
                                                 unsigned lds_byte_off,
                                                 unsigned rows_avail) {
  const unsigned long long ga = (unsigned long long)(uintptr_t)gsrc;
  u32x4 g0;
  g0[0] = 1u;                                                 // count=1, user mode
  g0[1] = lds_byte_off;                                       // lds_addr (bytes)
  g0[2] = (unsigned)(ga & 0xFFFFFFFFu);                       // global_addr[31:0]
  g0[3] = (unsigned)((ga >> 32) & 0x01FFFFFFu) | (2u << 30);  // addr[56:32] | type=2

  i32x8 g1;
  g1[0] = (int)((2u << 16) | (1u << 20) | (6u << 22) | (3u << 25)); // dsize/pad cfg
  g1[1] = (int)(128u << 16);                                  // tensor_dim0 = 128 (lo16)
  g1[2] = (int)((rows_avail & 0xFFFFu) << 16);                // dim0 hi16=0 | tensor_dim1 lo16
  g1[3] = (int)(((rows_avail >> 16) & 0xFFFFu) | (128u << 16)); // dim1 hi16 | tile_dim0=128
  g1[4] = (int)128u;                                          // tile_dim1=128, tile_dim2=0
  g1[5] = (int)128u;                                          // tensor_dim0_stride lo32 = 128
  g1[6] = 0;                                                  // stride0 hi16 | stride1 lo16
  g1[7] = 0;                                                  // stride1 hi32
  const i32x4 z4 = {0, 0, 0, 0};                              // 2-D tensor: groups 2/3 unused
  const i32x8 z8 = {0, 0, 0, 0, 0, 0, 0, 0};
  __builtin_amdgcn_tensor_load_to_lds(g0, g1, z4, z4, z8, 0);
}

// ---------------------------------------------------------------------------
// Fused GEMM: Y[M,128] = epilogue(X[M,128] @ W[128,128])
// epilogue: (+bias) -> (BN eval: gamma*(v*BN_INV)+beta) -> (ReLU)
// Block = 8 waves, 128-row tile; wave: 16 rows x full 128 cols.
// A tile staged by TDM (padded + zero-filled tail); W staged as K-pairs so
// each B fragment is one ds_load_b64. fp32 WMMA 16x16x4, 32 K-steps.
// ---------------------------------------------------------------------------
__launch_bounds__(256)
__global__ void k_gemm128(const float* __restrict__ X, const float* __restrict__ W,
                          const float* __restrict__ bias, const float* __restrict__ gamma,
                          const float* __restrict__ beta, float* __restrict__ Y,
                          int M, int relu) {
  __shared__ float As[128 * LSTRIDE];        // A tile, 132-float pitch (TDM pad)
  __shared__ v2f   Wp[64 * PSTRIDE];         // Wp[p*PSTRIDE+c] = {W[2p][c], W[2p+1][c]}

  const int tid  = threadIdx.x;
  const int wave = tid >> 5;
  const int lane = tid & 31;
  const int rbase = blockIdx.x * 128;

  // ---- A tile: one TDM issue from wave 0 (EXEC ignored; branch is uniform) ----
  if (wave == 0) {
    tdm_load_tile128(X + (size_t)rbase * HIDC,
                     (unsigned)(uintptr_t)(&As[0]),
                     (unsigned)(M - rbase));
  }

  // ---- W: stage interleaved K-pairs (overlaps with the TDM transfer) ----
  {
    const int c4 = (tid & 31) * 4;
    const int p0 = tid >> 5;                 // 0..7
#pragma unroll
    for (int i = 0; i < 8; ++i) {
      const int p = p0 + i * 8;              // pair row 0..63  (K = 2p, 2p+1)
      const float4 e = *(const float4*)(W + (size_t)(2 * p + 0) * HIDC + c4);
      const float4 o = *(const float4*)(W + (size_t)(2 * p + 1) * HIDC + c4);
      float4* dstv = (float4*)(&Wp[p * PSTRIDE + c4]);
      dstv[0] = make_float4(e.x, o.x, e.y, o.y);
      dstv[1] = make_float4(e.z, o.z, e.w, o.w);
    }
  }
  __builtin_amdgcn_s_wait_tensorcnt(0);      // no-op for waves with TENSORcnt==0
  __syncthreads();

  const int hf   = lane >> 4;        // half-wave selects K sub-pair (ISA 7.12.2 layout)
  const int l16  = lane & 15;
  const int arow = wave * 16 + l16;  // M index within tile (same for both halves)

  v8f acc[8] = {};

#pragma unroll 4
  for (int kk = 0; kk < 32; ++kk) {
    const int kb = kk * 4 + hf * 2;          // even -> 8B-aligned A read
    const v2f a = *(const v2f*)(&As[arow * LSTRIDE + kb]);
    const int pr = (2 * kk + hf) * PSTRIDE;  // pair row for this half-wave
#pragma unroll
    for (int n = 0; n < 8; ++n) {
      const v2f b = Wp[pr + n * 16 + l16];   // single ds_load_b64, pair-aligned
      acc[n] = __builtin_amdgcn_wmma_f32_16x16x4_f32(false, a, false, b,
                                                     (short)0, acc[n], false, false);
    }
  }

  // ---- Epilogue + store. C/D: VGPR r -> M=r (+8 for lanes 16..31), N = lane&15. ----
#pragma unroll
  for (int n = 0; n < 8; ++n) {
    const int col = n * 16 + l16;
    const float badd = bias  ? bias[col]  : 0.0f;
    const float g    = gamma ? gamma[col] : 0.0f;
    const float be   = gamma ? beta[col]  : 0.0f;
#pragma unroll
    for (int r = 0; r < 8; ++r) {
      const int row = rbase + wave * 16 + hf * 8 + r;
      if (row < M) {
        float v = acc[n][r] + badd;
        if (gamma) v = g * (v * BN_INV) + be;
        if (relu)  v = fmaxf(v, 0.0f);
        Y[(size_t)row * HIDC + col] = v;
      }
    }
  }
}

// ---------------------------------------------------------------------------
// Degree / normalization prep
// ---------------------------------------------------------------------------
__global__ void k_deg_init(float* deg, int N) {
  int i = blockIdx.x * blockDim.x + threadIdx.x;
  if (i < N) deg[i] = 1.0f;                       // self-loop contribution
}
__global__ void k_deg_add(float* deg, const int* __restrict__ dst, int E) {
  int i = blockIdx.x * blockDim.x + threadIdx.x;
  if (i < E) atomAddF(&deg[dst[i]], 1.0f);
}
__global__ void k_rsqrt(float* a, int N) {
  int i = blockIdx.x * blockDim.x + threadIdx.x;
  if (i < N) a[i] = rsqrtf(a[i]);
}

// ---------------------------------------------------------------------------
// SpMM scatter: one wave per (edge or self-loop), 128 ch = 4 floats/lane.
// agg[dst] += m[src] * dinv[src]*dinv[dst]
// ---------------------------------------------------------------------------
__launch_bounds__(256)
__global__ void k_spmm(const float* __restrict__ m, const int* __restrict__ src,
                       const int* __restrict__ dst, const float* __restrict__ dinv,
                       float* __restrict__ agg, int E, int N) {
  const int g    = blockIdx.x * (blockDim.x >> 5) + (threadIdx.x >> 5);
  const int lane = threadIdx.x & 31;
  if (g >= E + N) return;
  int s, d; float w;
  if (g < E) { s = src[g]; d = dst[g]; w = dinv[s] * dinv[d]; }
  else       { s = g - E;  d = s;      float di = dinv[s]; w = di * di; }
  const float4 mv = *(const float4*)(m + (size_t)s * HIDC + lane * 4);
  float* ap = agg + (size_t)d * HIDC + lane * 4;
  atomAddF(ap + 0, mv.x * w);
  atomAddF(ap + 1, mv.y * w);
  atomAddF(ap + 2, mv.z * w);
  atomAddF(ap + 3, mv.w * w);
}

// ---------------------------------------------------------------------------
// Elementwise helpers (float4, grid-stride)
// ---------------------------------------------------------------------------
__global__ void k_zero(float* p, size_t n4) {
  size_t i = (size_t)blockIdx.x * blockDim.x + threadIdx.x;
  for (; i < n4; i += (size_t)gridDim.x * blockDim.x)
    ((float4*)p)[i] = make_float4(0.f, 0.f, 0.f, 0.f);
}
__global__ void k_max(const float* __restrict__ a, const float* __restrict__ b,
                      float* __restrict__ o, size_t n4) {
  size_t i = (size_t)blockIdx.x * blockDim.x + threadIdx.x;
  for (; i < n4; i += (size_t)gridDim.x * blockDim.x) {
    float4 va = ((const float4*)a)[i], vb = ((const float4*)b)[i];
    float4 vo = make_float4(fmaxf(va.x, vb.x), fmaxf(va.y, vb.y),
                            fmaxf(va.z, vb.z), fmaxf(va.w, vb.w));
    ((float4*)o)[i] = vo;
  }
}

extern "C" void kernel_launch(void* const* d_in, const int* in_sizes, int n_in,
                              void* d_out, int out_size, void* d_ws, size_t ws_size,
                              hipStream_t stream) {
  const float* x        = (const float*)d_in[0];
  const int*   eidx     = (const int*)  d_in[1];
  const float* fc0_w    = (const float*)d_in[2];
  const float* fc0_b    = (const float*)d_in[3];
  const float* conv_w   = (const float*)d_in[4];   // [2,128,128]
  const float* W_w      = (const float*)d_in[5];
  const float* W_b      = (const float*)d_in[6];
  const float* bn_gamma = (const float*)d_in[7];   // [3,128]
  const float* bn_beta  = (const float*)d_in[8];
  const float* fc_out_w = (const float*)d_in[9];
  const float* fc_out_b = (const float*)d_in[10];

  const int N = in_sizes[0] / HIDC;
  const int E = in_sizes[1] / 2;
  const int* src = eidx;
  const int* dst = eidx + E;

  float* ws = (float*)d_ws;
  const size_t Npad = ((size_t)N + 255) & ~(size_t)255;
  const size_t FEAT = (size_t)N * HIDC;
  float* dinv = ws;            // [N]
  float* bufA = ws + Npad;     // h0, then running JK-max
  float* bufB = bufA + FEAT;   // h1
  float* bufD = bufB + FEAT;   // agg scratch
  float* out  = (float*)d_out; // also used for m1/m2/h2/jk (in-place safe)

  const dim3 blk(256);
  const dim3 gN((N + 255) / 256), gE((E + 255) / 256);
  const dim3 gG((N + 127) / 128);                       // GEMM tiles
  const dim3 gS(((E + N) + 7) / 8);                     // 8 waves/block -> 8 edges/block
  const dim3 gZ(2048);
  const size_t n4 = FEAT / 4;

  // Degree + symmetric normalization
  k_deg_init<<<gN, blk, 0, stream>>>(dinv, N);
  k_deg_add <<<gE, blk, 0, stream>>>(dinv, dst, E);
  k_rsqrt   <<<gN, blk, 0, stream>>>(dinv, N);

  // h0 = relu(bn0(x @ fc0_w + fc0_b))
  k_gemm128<<<gG, blk, 0, stream>>>(x, fc0_w, fc0_b, bn_gamma + 0, bn_beta + 0, bufA, N, 1);

  // ---- layer 1 ----
  k_gemm128<<<gG, blk, 0, stream>>>(bufA, conv_w, nullptr, nullptr, nullptr, out, N, 0); // m1
  k_zero   <<<gZ, blk, 0, stream>>>(bufD, n4);
  k_spmm   <<<gS, blk, 0, stream>>>(out, src, dst, dinv, bufD, E, N);                    // agg1
  k_gemm128<<<gG, blk, 0, stream>>>(bufD, W_w, W_b, bn_gamma + HIDC, bn_beta + HIDC, bufB, N, 1); // h1
  k_max    <<<gZ, blk, 0, stream>>>(bufA, bufB, bufA, n4);                               // jk partial

  // ---- layer 2 ----
  k_gemm128<<<gG, blk, 0, stream>>>(bufB, conv_w + HIDC * HIDC, nullptr, nullptr, nullptr, out, N, 0); // m2
  k_zero   <<<gZ, blk, 0, stream>>>(bufD, n4);
  k_spmm   <<<gS, blk, 0, stream>>>(out, src, dst, dinv, bufD, E, N);                    // agg2
  k_gemm128<<<gG, blk, 0, stream>>>(bufD, W_w, W_b, bn_gamma + 2 * HIDC, bn_beta + 2 * HIDC, out, N, 1); // h2
  k_max    <<<gZ, blk, 0, stream>>>(bufA, out, out, n4);                                 // jk

  // out = jk @ fc_out_w + fc_out_b   (in-place: each block stages its own rows in LDS first)
  k_gemm128<<<gG, blk, 0, stream>>>(out, fc_out_w, fc_out_b, nullptr, nullptr, out, N, 0);
}